// ColonFormerLoss_90005334655687
// MI455X (gfx1250) — compile-verified
//
#include <hip/hip_runtime.h>
#include <math.h>

typedef __attribute__((ext_vector_type(2))) float v2f;
typedef __attribute__((ext_vector_type(8))) float v8f;

#define BB 8
#define HH 256
#define WW 256
#define NIMG (HH * WW)     // 65536
#define NTOT (BB * NIMG)   // 524288
#define LARGE_F 1.0e6f

// ---------------------------------------------------------------------------
// Wave32 sum via V_WMMA_F32_16X16X4_F32.
// A (16x4 f32) layout (ISA 7.12.2): lane m<16 supplies A[m][0..1] in vgpr0/1,
// lane m+16 supplies A[m][2..3].  With a1=0, rowsum_m = x[m] + x[m+16].
// B = all-ones (layout-independent).  D vgpr r: lanes0-15 = rowsum_r,
// lanes16-31 = rowsum_{r+8}.  Sum d[0..7] gives a half sum; shfl_xor(16)
// folds halves -> full 32-lane sum broadcast to every lane.
// ---------------------------------------------------------------------------
__device__ __forceinline__ float wave_sum32(float x) {
  v2f a; a[0] = x;    a[1] = 0.0f;
  v2f b; b[0] = 1.0f; b[1] = 1.0f;
  v8f c = {};
  v8f d = __builtin_amdgcn_wmma_f32_16x16x4_f32(false, a, false, b,
                                                (short)0, c, false, false);
  float s = d[0] + d[1] + d[2] + d[3] + d[4] + d[5] + d[6] + d[7];
  s += __shfl_xor(s, 16, 32);
  return s;
}

// K0: zero the stats/accumulator region (must happen every launch).
__global__ void k0_zero(unsigned* __restrict__ stats, int n) {
  int i = blockIdx.x * blockDim.x + threadIdx.x;
  if (i < n) stats[i] = 0u;
}

// K1: column pass of the EDT for both masks (feat0: t==0 -> "fg" EDT,
// feat1: t==1 -> "bg" EDT).  One thread per (b,w) column; forward scan
// stores distance-to-nearest-true-above, backward scan folds in
// distance-below, clamps at LARGE, squares.  Coalesced along w.
__global__ __launch_bounds__(256) void k1_col(const int* __restrict__ tg,
                                              float* __restrict__ buf0,
                                              float* __restrict__ buf1) {
  int c = blockIdx.x * blockDim.x + threadIdx.x;   // 0..2047
  int b = c >> 8;
  int w = c & (WW - 1);
  int base = b * NIMG + w;

  float last0 = -1.0e7f, last1 = -1.0e7f;
  for (int h = 0; h < HH; ++h) {
    int idx = base + h * WW;
    int t = tg[idx];
    float fh = (float)h;
    if (t == 0) last0 = fh; else last1 = fh;
    buf0[idx] = fh - last0;
    buf1[idx] = fh - last1;
  }
  float nxt0 = 1.0e7f, nxt1 = 1.0e7f;
  for (int h = HH - 1; h >= 0; --h) {
    int idx = base + h * WW;
    int t = tg[idx];
    float fh = (float)h;
    if (t == 0) nxt0 = fh; else nxt1 = fh;
    float g0 = fminf(fminf(buf0[idx], nxt0 - fh), LARGE_F);
    float g1 = fminf(fminf(buf1[idx], nxt1 - fh), LARGE_F);
    buf0[idx] = g0 * g0;
    buf1[idx] = g1 * g1;
  }
}

// K2: row pass: d2[j] = min_k(gsq[k] + (j-k)^2) for both masks from LDS,
// produce signed distance (fg where t==1 else -bg), overwrite buf0 with it,
// and fold per-image max|dist| + has_fg via one atomic per block.
__global__ __launch_bounds__(256) void k2_row(const int* __restrict__ tg,
                                              float* __restrict__ buf0,
                                              const float* __restrict__ buf1,
                                              unsigned* __restrict__ maxAbsBits,
                                              int* __restrict__ hasFg) {
  __shared__ float g0[WW];
  __shared__ float g1[WW];
  __shared__ float smax[WW];
  __shared__ int sany;

  int row = blockIdx.x;            // b*H + h
  int b = row >> 8;
  int j = threadIdx.x;
  int idx = row * WW + j;

  g0[j] = buf0[idx];
  g1[j] = buf1[idx];
  if (j == 0) sany = 0;
  __syncthreads();

  float m0 = 1.0e30f, m1 = 1.0e30f;
#pragma unroll 8
  for (int k = 0; k < WW; ++k) {
    float fd = (float)(j - k);
    float c2 = fd * fd;
    m0 = fminf(m0, g0[k] + c2);
    m1 = fminf(m1, g1[k] + c2);
  }
  float fg = sqrtf(m0);
  float bg = sqrtf(m1);
  int t = tg[idx];
  float dist = t ? fg : -bg;
  buf0[idx] = dist;                 // LDS already holds gsq; safe overwrite
  if (t) sany = 1;                  // benign race, all writers store 1

  smax[j] = fabsf(dist);
  __syncthreads();
  for (int s = WW / 2; s > 0; s >>= 1) {
    if (j < s) smax[j] = fmaxf(smax[j], smax[j + s]);
    __syncthreads();
  }
  if (j == 0) {
    atomicMax(&maxAbsBits[b], __float_as_uint(smax[0]));  // |dist|>=0: bit-ordered
    if (sany) atomicOr(&hasFg[b], 1);
  }
}

// K3: fused focal + weighted-IoU accumulation over all 4 predictions.
// Grid = B images x 8 chunks; 12 partial sums per thread, reduced across the
// wave with 12 f32 WMMAs, then across 8 waves in LDS, then atomicAdd.
__global__ __launch_bounds__(256) void k3_loss(const float* __restrict__ p0,
                                               const float* __restrict__ p1,
                                               const float* __restrict__ p2,
                                               const float* __restrict__ p3,
                                               const int* __restrict__ tg,
                                               const float* __restrict__ dist,
                                               const unsigned* __restrict__ maxAbsBits,
                                               const int* __restrict__ hasFg,
                                               float* __restrict__ focalG,
                                               float* __restrict__ interG,
                                               float* __restrict__ uniG) {
  const int CH = 8;
  const int seg = NIMG / CH;       // 8192 elements per block
  int b = blockIdx.x / CH;
  int c = blockIdx.x % CH;
  int base = b * NIMG + c * seg;

  float md = __uint_as_float(maxAbsBits[b]);
  int valid = hasFg[b] && (md > 0.0f);
  float inv_md3 = 3.0f / fmaxf(md, 1e-12f);

  const float* preds[4] = {p0, p1, p2, p3};
  float facc[4] = {0.f, 0.f, 0.f, 0.f};
  float iacc[4] = {0.f, 0.f, 0.f, 0.f};
  float uacc[4] = {0.f, 0.f, 0.f, 0.f};

  for (int i = threadIdx.x; i < seg; i += 256) {
    int idx = base + i;
    float tf = (tg[idx] != 0) ? 1.0f : 0.0f;
    float dd = dist[idx];
    float w = valid ? (1.0f + expf(-fabsf(dd) * inv_md3)) : 1.0f;
#pragma unroll
    for (int p = 0; p < 4; ++p) {
      float x = preds[p][idx];
      float sig = 1.0f / (1.0f + expf(-x));
      float spp = (x < -20.f) ? -x : log1pf(expf(-x));  // softplus(-x)
      float spn = (x > 20.f) ? x : log1pf(expf(x));     // softplus(x)
      float ce = tf * spp + (1.0f - tf) * spn;
      float p_t = sig * tf + (1.0f - sig) * (1.0f - tf);
      float omp = 1.0f - p_t;
      float at = 0.25f * tf + 0.75f * (1.0f - tf);
      facc[p] += at * omp * omp * ce;
      iacc[p] += sig * tf * w;
      uacc[p] += (sig + tf) * w;
    }
  }

  __shared__ float sw[8][12];
  int wave = threadIdx.x >> 5;
  int lane = threadIdx.x & 31;
  float r[12];
#pragma unroll
  for (int p = 0; p < 4; ++p) { r[p] = facc[p]; r[4 + p] = iacc[p]; r[8 + p] = uacc[p]; }
#pragma unroll
  for (int q = 0; q < 12; ++q) {
    float s = wave_sum32(r[q]);          // v_wmma_f32_16x16x4_f32 path
    if (lane == 0) sw[wave][q] = s;
  }
  __syncthreads();
  if (threadIdx.x < 12) {
    float s = 0.0f;
    for (int wv = 0; wv < 8; ++wv) s += sw[wv][threadIdx.x];
    int q = threadIdx.x;
    if (q < 4)      atomicAdd(&focalG[q], s);
    else if (q < 8) atomicAdd(&interG[(q - 4) * BB + b], s);
    else            atomicAdd(&uniG[(q - 8) * BB + b], s);
  }
}

// K4: scalar combine (68 inputs -> 1 output).
__global__ void k4_final(const float* __restrict__ focalG,
                         const float* __restrict__ interG,
                         const float* __restrict__ uniG,
                         float* __restrict__ out) {
  if (threadIdx.x == 0 && blockIdx.x == 0) {
    const float wt[4] = {1.0f, 0.4f, 0.2f, 0.4f / 3.0f};
    float total = 0.0f;
    for (int p = 0; p < 4; ++p) {
      float fl = focalG[p] / (float)NTOT;
      float ious = 0.0f;
      for (int b = 0; b < BB; ++b) {
        float inter = interG[p * BB + b];
        float uni = uniG[p * BB + b] - inter;
        ious += (inter + 1e-6f) / (uni + 1e-6f);
      }
      total += wt[p] * (fl + (1.0f - ious / (float)BB));
    }
    *out = total;
  }
}

extern "C" void kernel_launch(void* const* d_in, const int* in_sizes, int n_in,
                              void* d_out, int out_size, void* d_ws, size_t ws_size,
                              hipStream_t stream) {
  const float* p0 = (const float*)d_in[0];
  const float* p1 = (const float*)d_in[1];
  const float* p2 = (const float*)d_in[2];
  const float* p3 = (const float*)d_in[3];
  const int*   tg = (const int*)d_in[4];

  float*    buf0       = (float*)d_ws;                 // NTOT: gsq_fg -> dist
  float*    buf1       = buf0 + NTOT;                  // NTOT: gsq_bg
  unsigned* maxAbsBits = (unsigned*)(buf1 + NTOT);     // BB
  int*      hasFg      = (int*)(maxAbsBits + BB);      // BB
  float*    focalG     = (float*)(hasFg + BB);         // 4
  float*    interG     = focalG + 4;                   // 4*BB
  float*    uniG       = interG + 4 * BB;              // 4*BB

  const int nStats = BB + BB + 4 + 4 * BB + 4 * BB;    // 84 words
  k0_zero<<<1, 128, 0, stream>>>(maxAbsBits, nStats);
  k1_col<<<(BB * WW) / 256, 256, 0, stream>>>(tg, buf0, buf1);
  k2_row<<<BB * HH, WW, 0, stream>>>(tg, buf0, buf1, maxAbsBits, hasFg);
  k3_loss<<<BB * 8, 256, 0, stream>>>(p0, p1, p2, p3, tg, buf0,
                                      maxAbsBits, hasFg, focalG, interG, uniG);
  k4_final<<<1, 32, 0, stream>>>(focalG, interG, uniG, (float*)d_out);
}